// VA_2113123910298
// MI455X (gfx1250) — compile-verified
//
#include <hip/hip_runtime.h>
#include <math.h>

// ---------------------------------------------------------------------------
// va_forward for MI455X (gfx1250):
//   per row n (of B*T): q = (p + t) * (p != 0) elementwise, out = R(angles) @ q
// Memory-bound: 383 MB traffic, ~0.26 GFLOP -> design for 23.3 TB/s streaming.
// CDNA5 paths used:
//   * GLOBAL_LOAD_ASYNC_TO_LDS_B128 / GLOBAL_STORE_ASYNC_FROM_LDS_B128 (ASYNCcnt)
//   * s_wait_asynccnt
//   * V_WMMA_F32_16X16X4_F32: per-row homogeneous transform as a K=4 f32 GEMM,
//     A = [q | 1] (16x4), B = [R^T ; 0] (4x16). Offloads the matvec FMAs to the
//     matrix pipe; streaming structure unchanged.
// ---------------------------------------------------------------------------

#define VA_D        75                 // floats per row (25 points * 3)
#define VA_J        25                 // points per row
#define VA_ROWS     32                 // rows per block (== waves per block)
#define VA_TILE     (VA_ROWS * VA_D)   // 2400 floats = 9600 B (16B multiple)
#define VA_THT      (VA_ROWS * 6)      // 192 floats  = 768 B  (16B multiple)
#define VA_THREADS  1024               // 32 waves; 1 wave per row in phase 3
#define VA_RSTRIDE  13                 // padded R row (9 R + 3 t), odd -> no bank conflicts

#if __has_builtin(__builtin_amdgcn_wmma_f32_16x16x4_f32)
#define VA_HAVE_WMMA 1
typedef __attribute__((ext_vector_type(2))) float v2f;
typedef __attribute__((ext_vector_type(8))) float v8f;
#endif

// Generic (flat) pointers to LDS carry the wave-relative LDS byte offset in
// their low 32 bits (aperture bits live in the high half; HW uses addr[31:0]).
static __device__ __forceinline__ unsigned lds_addr_of(const void* p) {
  return (unsigned)(unsigned long long)p;
}

// CDNA5 async copy: memory -> LDS, 16 bytes per active lane, tracked by ASYNCcnt.
static __device__ __forceinline__ void async_g2l_b128(unsigned lds, const float* g) {
  asm volatile("global_load_async_to_lds_b128 %0, %1, off"
               :: "v"(lds), "v"(g)
               : "memory");
}
// CDNA5 async copy: LDS -> memory, 16 bytes per active lane.
static __device__ __forceinline__ void async_l2g_b128(float* g, unsigned lds) {
  asm volatile("global_store_async_from_lds_b128 %0, %1, off"
               :: "v"(g), "v"(lds)
               : "memory");
}
static __device__ __forceinline__ void wait_async0() {
#if __has_builtin(__builtin_amdgcn_s_wait_asynccnt)
  __builtin_amdgcn_s_wait_asynccnt(0);
#else
  asm volatile("s_wait_asynccnt 0" ::: "memory");
#endif
}

__global__ __launch_bounds__(VA_THREADS, 1)
void va_forward_tile(const float* __restrict__ conv,
                     const float* __restrict__ theta,
                     float* __restrict__ out)
{
  __shared__ __align__(16) float s_tile[VA_TILE];           // in-place point tile
  __shared__ __align__(16) float s_th[VA_THT];              // theta tile
  __shared__ float s_R[VA_ROWS * VA_RSTRIDE];               // per-row R (9) + t (3)

  const int t = threadIdx.x;
  const long long row0 = (long long)blockIdx.x * VA_ROWS;
  const float* gconv = conv  + row0 * VA_D;
  const float* gth   = theta + row0 * 6;
  float*       gout  = out   + row0 * VA_D;

  // ---- Phase 1: async-stage tiles into LDS (B128 per lane) ----------------
  if (t < VA_TILE / 4) {                               // 600 lanes: conv tile
    async_g2l_b128(lds_addr_of(&s_tile[t * 4]), gconv + t * 4);
  } else if (t < VA_TILE / 4 + VA_THT / 4) {           // 48 lanes: theta tile
    const int i = t - VA_TILE / 4;
    async_g2l_b128(lds_addr_of(&s_th[i * 4]), gth + i * 4);
  }
  wait_async0();
  __syncthreads();

  // ---- Phase 2: wave 0 (dense 32 lanes) builds R = Rz(c)·Ry(a)·Rx(b) ------
  // Amortizes 6 transcendentals per ROW (not per point); redundant per-point
  // sincos would cost ~3x the 16.4us memory floor in VALU time.
  if (t < VA_ROWS) {
    const float a = s_th[t * 6 + 0];   // ry angle
    const float b = s_th[t * 6 + 1];   // rx angle
    const float c = s_th[t * 6 + 2];   // rz angle
    float sa, ca, sb, cb, sc, cc;
    sincosf(a, &sa, &ca);
    sincosf(b, &sb, &cb);
    sincosf(c, &sc, &cc);
    float* R = &s_R[t * VA_RSTRIDE];
    R[0] = cc * ca;  R[1] = cc * sa * sb - sc * cb;  R[2] = cc * sa * cb + sc * sb;
    R[3] = sc * ca;  R[4] = sc * sa * sb + cc * cb;  R[5] = sc * sa * cb - cc * sb;
    R[6] = -sa;      R[7] = ca * sb;                 R[8] = ca * cb;
    R[9]  = s_th[t * 6 + 3];   // tx
    R[10] = s_th[t * 6 + 4];   // ty
    R[11] = s_th[t * 6 + 5];   // tz
  }
  __syncthreads();

  // ---- Phase 3: one wave per row; transform via V_WMMA_F32_16X16X4_F32 ----
  // Runs unconditionally on all 32 waves -> EXEC is all-ones at the WMMA.
  {
    const int w    = t >> 5;           // wave id == row within tile
    const int lane = t & 31;
    const int n16  = lane & 15;
    const bool hiK = lane >= 16;       // A: K=2,3 half; D: M+8 half
    const float* Rw = &s_R[w * VA_RSTRIDE];
    const float tx = Rw[9], ty = Rw[10], tz = Rw[11];

#ifdef VA_HAVE_WMMA
    // B (4x16): lanes 0-15 carry K=0,1 for column n16; lanes 16-31 carry K=2,3.
    // B[k][n] = R[n][k] for n<3, else 0; K=3 row = 0 (translation already in q).
    float b0 = 0.0f, b1 = 0.0f;
    if (n16 < 3) {
      b0 = hiK ? Rw[n16 * 3 + 2] : Rw[n16 * 3 + 0];
      b1 = hiK ? 0.0f            : Rw[n16 * 3 + 1];
    }
    const v2f B = {b0, b1};

#pragma unroll
    for (int m0 = 0; m0 < 32; m0 += 16) {
      // A (16x4): lane M = n16; lanes 0-15 hold (qx,qy), lanes 16-31 hold (qz,1).
      const int pt  = m0 + n16;
      const int ptc = pt > (VA_J - 1) ? (VA_J - 1) : pt;   // clamp pad rows (dup reads)
      const int base = w * VA_D + ptc * 3;
      float a0, a1;
      if (!hiK) {
        const float x = s_tile[base + 0];
        const float y = s_tile[base + 1];
        a0 = (x != 0.0f) ? (x + tx) : 0.0f;
        a1 = (y != 0.0f) ? (y + ty) : 0.0f;
      } else {
        const float z = s_tile[base + 2];
        a0 = (z != 0.0f) ? (z + tz) : 0.0f;
        a1 = 1.0f;                                          // homogeneous coord
      }
      const v2f A = {a0, a1};
      v8f C = {};
      // 8 args: (neg_a, A, neg_b, B, c_mod, C, reuse_a, reuse_b)
      C = __builtin_amdgcn_wmma_f32_16x16x4_f32(false, A, false, B,
                                                (short)0, C, false, false);
      // Scatter D: lane n (n<3) holds coord n of points m0+v (VGPR v); the
      // hiK half holds points m0+8+v.  In place: iter m0=16 reads pts 16-24,
      // never written by iter m0=0 (which writes pts 0-15).
      if (n16 < 3) {
        const int ptBase = m0 + (hiK ? 8 : 0);
#pragma unroll
        for (int v = 0; v < 8; ++v) {
          const int p2 = ptBase + v;
          if (p2 < VA_J) s_tile[w * VA_D + p2 * 3 + n16] = C[v];
        }
      }
    }
#else
    // Scalar fallback (same math), point-per-thread over the wave's row.
    for (int p = lane; p < VA_J; p += 32) {
      float* q = &s_tile[w * VA_D + p * 3];
      const float x = q[0], y = q[1], z = q[2];
      const float qx = (x != 0.0f) ? (x + tx) : 0.0f;
      const float qy = (y != 0.0f) ? (y + ty) : 0.0f;
      const float qz = (z != 0.0f) ? (z + tz) : 0.0f;
      q[0] = Rw[0] * qx + Rw[1] * qy + Rw[2] * qz;
      q[1] = Rw[3] * qx + Rw[4] * qy + Rw[5] * qz;
      q[2] = Rw[6] * qx + Rw[7] * qy + Rw[8] * qz;
    }
#endif
  }
  __syncthreads();

  // ---- Phase 4: async B128 store LDS -> global ----------------------------
  if (t < VA_TILE / 4) {
    async_l2g_b128(gout + t * 4, lds_addr_of(&s_tile[t * 4]));
  }
  wait_async0();  // S_ENDPGM also does an implicit wait-idle
}

// Scalar tail kernel for row counts not divisible by VA_ROWS (robustness;
// the harness shape 2048*300 = 614400 divides exactly, so this never runs).
__global__ void va_forward_tail(const float* __restrict__ conv,
                                const float* __restrict__ theta,
                                float* __restrict__ out,
                                long long row_begin, long long row_end)
{
  const long long pid = row_begin * VA_J +
                        (long long)blockIdx.x * blockDim.x + threadIdx.x;
  const long long row = pid / VA_J;
  if (row >= row_end) return;
  const int p = (int)(pid % VA_J);

  const float* th = theta + row * 6;
  float sa, ca, sb, cb, sc, cc;
  sincosf(th[0], &sa, &ca);
  sincosf(th[1], &sb, &cb);
  sincosf(th[2], &sc, &cc);
  const float R0 = cc * ca, R1 = cc * sa * sb - sc * cb, R2 = cc * sa * cb + sc * sb;
  const float R3 = sc * ca, R4 = sc * sa * sb + cc * cb, R5 = sc * sa * cb - cc * sb;
  const float R6 = -sa,     R7 = ca * sb,                R8 = ca * cb;

  const float* g = conv + row * VA_D + p * 3;
  float*       o = out  + row * VA_D + p * 3;
  const float x = g[0], y = g[1], z = g[2];
  const float qx = (x != 0.0f) ? (x + th[3]) : 0.0f;
  const float qy = (y != 0.0f) ? (y + th[4]) : 0.0f;
  const float qz = (z != 0.0f) ? (z + th[5]) : 0.0f;
  o[0] = R0 * qx + R1 * qy + R2 * qz;
  o[1] = R3 * qx + R4 * qy + R5 * qz;
  o[2] = R6 * qx + R7 * qy + R8 * qz;
}

extern "C" void kernel_launch(void* const* d_in, const int* in_sizes, int n_in,
                              void* d_out, int out_size, void* d_ws, size_t ws_size,
                              hipStream_t stream) {
  const float* conv  = (const float*)d_in[0];   // (2048, 300, 75) f32
  const float* theta = (const float*)d_in[1];   // (2048, 300, 6)  f32
  float* out = (float*)d_out;                   // (2048, 300, 75) f32
  (void)d_ws; (void)ws_size; (void)n_in; (void)out_size;

  const long long nrows = (long long)in_sizes[0] / VA_D;   // 614400
  const long long full  = nrows / VA_ROWS;                  // 19200 blocks
  const long long tail  = nrows % VA_ROWS;                  // 0 for harness shapes

  if (full > 0) {
    hipLaunchKernelGGL(va_forward_tile, dim3((unsigned)full), dim3(VA_THREADS),
                       0, stream, conv, theta, out);
  }
  if (tail > 0) {
    const long long pts = tail * VA_J;
    hipLaunchKernelGGL(va_forward_tail,
                       dim3((unsigned)((pts + 255) / 256)), dim3(256),
                       0, stream, conv, theta, out, full * VA_ROWS, nrows);
  }
}